// SAGE_43593918054550
// MI455X (gfx1250) — compile-verified
//
#include <hip/hip_runtime.h>
#include <hip/hip_bf16.h>

#define N_NODES 100000
#define N_EDGES 600000
#define D 128
#define BN_EPS 1e-5f
#define NORM_EPS 1e-12f

typedef float v2f __attribute__((ext_vector_type(2)));
typedef float v8f __attribute__((ext_vector_type(8)));

// ---------------------------------------------------------------------------
// K0: h = BN(ReLU(x)) ; zero msg_sum ; zero deg.  One thread per float4.
// ---------------------------------------------------------------------------
__global__ void __launch_bounds__(256)
k_prep(const float* __restrict__ x,
       const float* __restrict__ gamma, const float* __restrict__ beta,
       const float* __restrict__ mean,  const float* __restrict__ var,
       float* __restrict__ h, float* __restrict__ msg, float* __restrict__ deg)
{
    const int tid = blockIdx.x * blockDim.x + threadIdx.x;
    if (tid < N_NODES) deg[tid] = 0.0f;

    const int total = N_NODES * (D / 4);
    if (tid >= total) return;

    const int c4 = (tid & (D / 4 - 1)) * 4;           // column of first component
    const float4 xv = reinterpret_cast<const float4*>(x)[tid];
    const float4 g  = *reinterpret_cast<const float4*>(gamma + c4);
    const float4 b  = *reinterpret_cast<const float4*>(beta  + c4);
    const float4 mu = *reinterpret_cast<const float4*>(mean  + c4);
    const float4 vv = *reinterpret_cast<const float4*>(var   + c4);

    float4 hv;
    {
        float s;
        s = g.x * rsqrtf(vv.x + BN_EPS); hv.x = fmaxf(xv.x, 0.0f) * s + (b.x - mu.x * s);
        s = g.y * rsqrtf(vv.y + BN_EPS); hv.y = fmaxf(xv.y, 0.0f) * s + (b.y - mu.y * s);
        s = g.z * rsqrtf(vv.z + BN_EPS); hv.z = fmaxf(xv.z, 0.0f) * s + (b.z - mu.z * s);
        s = g.w * rsqrtf(vv.w + BN_EPS); hv.w = fmaxf(xv.w, 0.0f) * s + (b.w - mu.w * s);
    }
    reinterpret_cast<float4*>(h)[tid]   = hv;
    reinterpret_cast<float4*>(msg)[tid] = make_float4(0.f, 0.f, 0.f, 0.f);
}

// ---------------------------------------------------------------------------
// K1: edge scatter.  One wave32 per edge: coalesced 512B gather of h[src],
//     f32 atomic scatter-add into msg_sum[dst]; lane 0 bumps deg[dst].
// ---------------------------------------------------------------------------
__global__ void __launch_bounds__(256)
k_scatter(const int* __restrict__ src, const int* __restrict__ dst,
          const float* __restrict__ h, float* __restrict__ msg,
          float* __restrict__ deg)
{
    const int edge = (blockIdx.x * blockDim.x + threadIdx.x) >> 5;
    const int lane = threadIdx.x & 31;
    if (edge >= N_EDGES) return;

    const int s = src[edge];
    const int d = dst[edge];

    const float4 v = *reinterpret_cast<const float4*>(h + (size_t)s * D + lane * 4);
    float* p = msg + (size_t)d * D + lane * 4;
    atomicAdd(p + 0, v.x);
    atomicAdd(p + 1, v.y);
    atomicAdd(p + 2, v.z);
    atomicAdd(p + 3, v.w);
    if (lane == 0) atomicAdd(deg + d, 1.0f);
}

// ---------------------------------------------------------------------------
// K2: out = normalize( (msg/deg) @ Wl^T + bl + h @ Wr^T ) via fp32 WMMA.
//     One wave per 16-node tile; 8 column tiles of 16; K=128 in steps of 4.
// ---------------------------------------------------------------------------
__global__ void __launch_bounds__(256)
k_gemm(const float* __restrict__ h,  const float* __restrict__ msg,
       const float* __restrict__ deg,
       const float* __restrict__ Wl, const float* __restrict__ bl,
       const float* __restrict__ Wr, float* __restrict__ out)
{
    const int tile = (blockIdx.x * blockDim.x + threadIdx.x) >> 5;
    if (tile >= N_NODES / 16) return;               // wave-uniform exit (EXEC all-1s inside)

    const int lane = threadIdx.x & 31;
    const int nl   = lane & 15;                     // A-row (M) and B-col (N) index
    const int hi   = lane >> 4;                     // K-pair select / C-row-half select
    const int base = tile * 16;
    const int row  = base + nl;

    const float invdeg = 1.0f / fmaxf(deg[row], 1.0f);
    const float* hrow  = h   + (size_t)row * D;
    const float* mrow  = msg + (size_t)row * D;

    v8f acc[8] = {};                                // 8 col-tiles x 8 VGPRs (f32 C/D)

    for (int kk = 0; kk < D; kk += 4) {
        const int k0 = kk + hi * 2;                 // ISA A/B layout: VGPR0=K0|K2, VGPR1=K1|K3
        v2f a_h = *reinterpret_cast<const v2f*>(hrow + k0);
        v2f a_m = *reinterpret_cast<const v2f*>(mrow + k0);
        a_m *= invdeg;                              // fold mean = msg/deg into A fragment

#pragma unroll
        for (int t = 0; t < 8; ++t) {
            const int n = t * 16 + nl;
            v2f b_l = *reinterpret_cast<const v2f*>(Wl + (size_t)n * D + k0);
            v2f b_r = *reinterpret_cast<const v2f*>(Wr + (size_t)n * D + k0);
            acc[t] = __builtin_amdgcn_wmma_f32_16x16x4_f32(
                         false, a_m, false, b_l, (short)0, acc[t], false, false);
            acc[t] = __builtin_amdgcn_wmma_f32_16x16x4_f32(
                         false, a_h, false, b_r, (short)0, acc[t], false, false);
        }
    }

    // Bias + per-row sum of squares (row r+8*hi lives across the 16 lanes of this half).
    float ssq[8] = {};
#pragma unroll
    for (int t = 0; t < 8; ++t) {
        const float bb = bl[t * 16 + nl];
#pragma unroll
        for (int r = 0; r < 8; ++r) {
            const float v = acc[t][r] + bb;
            acc[t][r] = v;
            ssq[r] += v * v;
        }
    }

    // Cross-lane reduce within each 16-lane half (masks 1,2,4,8 never cross halves).
#pragma unroll
    for (int r = 0; r < 8; ++r) {
        float s = ssq[r];
        s += __shfl_xor(s, 1);
        s += __shfl_xor(s, 2);
        s += __shfl_xor(s, 4);
        s += __shfl_xor(s, 8);
        ssq[r] = 1.0f / fmaxf(sqrtf(s), NORM_EPS);  // 1 / max(||row||, eps)
    }

    // Scale and store: lane writes (row = base + 8*hi + r, col = 16*t + nl).
#pragma unroll
    for (int t = 0; t < 8; ++t) {
#pragma unroll
        for (int r = 0; r < 8; ++r) {
            const int orow = base + hi * 8 + r;
            out[(size_t)orow * D + t * 16 + nl] = acc[t][r] * ssq[r];
        }
    }
}

// ---------------------------------------------------------------------------
extern "C" void kernel_launch(void* const* d_in, const int* in_sizes, int n_in,
                              void* d_out, int out_size, void* d_ws, size_t ws_size,
                              hipStream_t stream)
{
    const float* x     = (const float*)d_in[0];
    const int*   edge  = (const int*)  d_in[1];   // [2][N_EDGES]: src row then dst row
    // d_in[2..4] = Wl1/bl1/Wr1 : dead in the reference output — skipped entirely.
    const float* Wl2   = (const float*)d_in[5];
    const float* bl2   = (const float*)d_in[6];
    const float* Wr2   = (const float*)d_in[7];
    const float* gamma = (const float*)d_in[8];
    const float* beta  = (const float*)d_in[9];
    const float* mean  = (const float*)d_in[10];
    const float* var   = (const float*)d_in[11];
    float*       out   = (float*)d_out;

    float* h   = (float*)d_ws;                       // N*D f32
    float* msg = h   + (size_t)N_NODES * D;          // N*D f32
    float* deg = msg + (size_t)N_NODES * D;          // N   f32

    {   // K0: h = BN(ReLU(x)); zero accumulators
        const int threads = N_NODES * (D / 4);
        k_prep<<<(threads + 255) / 256, 256, 0, stream>>>(
            x, gamma, beta, mean, var, h, msg, deg);
    }
    {   // K1: edge scatter-add (one wave32 per edge)
        const long long threads = (long long)N_EDGES * 32;
        k_scatter<<<(int)((threads + 255) / 256), 256, 0, stream>>>(
            edge, edge + N_EDGES, h, msg, deg);
    }
    {   // K2: dual fp32-WMMA GEMM + bias + L2-normalize (one wave per 16 nodes)
        const long long threads = (long long)(N_NODES / 16) * 32;
        k_gemm<<<(int)((threads + 255) / 256), 256, 0, stream>>>(
            h, msg, deg, Wl2, bl2, Wr2, out);
    }
}